// Quanv2d_21217138442861
// MI455X (gfx1250) — compile-verified
//
#include <hip/hip_runtime.h>

typedef float v2f __attribute__((ext_vector_type(2)));
typedef float v8f __attribute__((ext_vector_type(8)));

#define IMG   128
#define PPD   63
#define PPB   (PPD * PPD)      /* 3969 patches per image           */
#define OUTB  (4 * PPB)        /* 15876 output floats per image    */

// Hardware transcendentals (v_sin_f32 / v_cos_f32). All arguments here are
// in [-2, 2] (inputs are uniform [0,1), weights likewise), so no range
// reduction is needed and HW accuracy is ample.
static __device__ __forceinline__ void fsincos(float x, float* s, float* c) {
    *s = __sinf(x);
    *c = __cosf(x);
}

// One wave (32 lanes) handles a tile of 16 patches.
//   lane&15  = p : A-matrix row (patch-in-tile)  AND  B-matrix column (output basis n)
//   lane>>4  = hi: selects which K-pair of each 4-wide WMMA chunk this lane supplies
// D = A x B accumulated over 4 chunks of K (K=16 total), once for Re, once for Im.
__global__ __launch_bounds__(128)
void quanv2d_wmma_kernel(const float* __restrict__ X,
                         const float* __restrict__ W,
                         float* __restrict__ out,
                         int nTiles, int nPatches)
{
    const int lane = threadIdx.x & 31;
    const int p    = lane & 15;
    const int hi   = lane >> 4;
    const int tile = blockIdx.x * 4 + (threadIdx.x >> 5);
    if (tile >= nTiles) return;                 // wave-uniform: EXEC stays all-ones for WMMA

    // ---------- patch coordinates for this lane's A-row ----------
    const int g  = tile * 16 + p;
    const int gl = (g < nPatches) ? g : (nPatches - 1);
    const unsigned b   = (unsigned)gl / PPB;
    const unsigned rm  = (unsigned)gl % PPB;
    const unsigned row = rm / PPD;
    const unsigned col = rm % PPD;

    // ---------- load the 4x4 patch (16 angles) ----------
    float ang[16];
    const float* px = X + ((size_t)b * IMG + row * 2) * IMG + col * 2;
    #pragma unroll
    for (int i = 0; i < 4; ++i)
        #pragma unroll
        for (int j = 0; j < 4; ++j)
            ang[i * 4 + j] = px[i * IMG + j];

    // ---------- single-qubit encoding chains: v[q] = (chain) H |0>  ----------
    const float IS2 = 0.70710678118654752f;
    float vr[4][2], vi[4][2];
    #pragma unroll
    for (int q = 0; q < 3; ++q) {
        float ur0 = IS2, ui0 = 0.f, ur1 = IS2, ui1 = 0.f;
        #pragma unroll
        for (int i = 0; i < 5; ++i) {
            float s, c; fsincos(0.5f * ang[q * 5 + i], &s, &c);
            float nr0, ni0, nr1, ni1;
            if ((i & 1) == 0) {                 // RZ: u0 *= e^{-i t/2}, u1 *= e^{+i t/2}
                nr0 = ur0 * c + ui0 * s;  ni0 = ui0 * c - ur0 * s;
                nr1 = ur1 * c - ui1 * s;  ni1 = ui1 * c + ur1 * s;
            } else {                            // RY: [[c,-s],[s,c]]
                nr0 = c * ur0 - s * ur1;  ni0 = c * ui0 - s * ui1;
                nr1 = s * ur0 + c * ur1;  ni1 = s * ui0 + c * ui1;
            }
            ur0 = nr0; ui0 = ni0; ur1 = nr1; ui1 = ni1;
        }
        vr[q][0] = ur0; vi[q][0] = ui0; vr[q][1] = ur1; vi[q][1] = ui1;
    }
    {   // qubit 3: only RZ(ang[15]) after H
        float s, c; fsincos(0.5f * ang[15], &s, &c);
        vr[3][0] = IS2 * c; vi[3][0] = -IS2 * s;
        vr[3][1] = IS2 * c; vi[3][1] =  IS2 * s;
    }

    // ---------- weights (uniform across grid) ----------
    float wth[4], cw[4], sw[4];
    #pragma unroll
    for (int i = 0; i < 4; ++i) wth[i] = W[i];
    #pragma unroll
    for (int q = 0; q < 4; ++q) fsincos(0.5f * W[4 + q], &sw[q], &cw[q]);

    // ---------- build WMMA operands ----------
    // A chunk kk, VGPR j: A[p][k],  k = 4*kk + 2*hi + j  (f32 16x4 A layout)
    // B chunk kk, VGPR j: B[k][p] = R[p][k] with R = ⊗q RY(w[4+q])  (mirrored 4x16 layout)
    v2f aRe[4], aIm[4], bM[4];
    const int nb0 = (p >> 3) & 1, nb1 = (p >> 2) & 1, nb2 = (p >> 1) & 1, nb3 = p & 1;
    #pragma unroll
    for (int kk = 0; kk < 4; ++kk) {
        #pragma unroll
        for (int j = 0; j < 2; ++j) {
            const int k  = kk * 4 + hi * 2 + j;
            const int k0 = (k >> 3) & 1, k1 = (k >> 2) & 1, k2 = (k >> 1) & 1, k3 = k & 1;
            // product-state amplitude s[k] = v0[k0]*v1[k1]*v2[k2]*v3[k3]
            float ar = vr[0][k0], ai = vi[0][k0];
            float tr = ar * vr[1][k1] - ai * vi[1][k1];
            float ti = ar * vi[1][k1] + ai * vr[1][k1];
            ar = tr * vr[2][k2] - ti * vi[2][k2];
            ai = tr * vi[2][k2] + ti * vr[2][k2];
            tr = ar * vr[3][k3] - ai * vi[3][k3];
            ti = ar * vi[3][k3] + ai * vr[3][k3];
            // CRZ ring = diagonal phase e^{i phi(k)}
            const int kb[4] = { k0, k1, k2, k3 };
            float phi = 0.f;
            #pragma unroll
            for (int i = 0; i < 4; ++i)
                phi += (float)kb[i] * wth[i] * ((float)kb[(i + 1) & 3] - 0.5f);
            float sp, cp; fsincos(phi, &sp, &cp);
            aRe[kk][j] = tr * cp - ti * sp;
            aIm[kk][j] = tr * sp + ti * cp;
            // B[k][n=p] = prod_q RY_q[n_q][k_q];  RY = [[c,-s],[s,c]]
            const int nbq[4] = { nb0, nb1, nb2, nb3 };
            float bv = 1.f;
            #pragma unroll
            for (int q = 0; q < 4; ++q)
                bv *= (nbq[q] == kb[q]) ? cw[q] : (kb[q] ? -sw[q] : sw[q]);
            bM[kk][j] = bv;
        }
    }

    // ---------- fixed 16x16 RY-layer transform on the matrix pipe ----------
    v8f accR = {0.f,0.f,0.f,0.f,0.f,0.f,0.f,0.f};
    v8f accI = {0.f,0.f,0.f,0.f,0.f,0.f,0.f,0.f};
    #pragma unroll
    for (int kk = 0; kk < 4; ++kk) {
        accR = __builtin_amdgcn_wmma_f32_16x16x4_f32(false, aRe[kk], false, bM[kk],
                                                     (short)0, accR, false, false);
        accI = __builtin_amdgcn_wmma_f32_16x16x4_f32(false, aIm[kk], false, bM[kk],
                                                     (short)0, accI, false, false);
    }

    // ---------- probabilities + Walsh-Hadamard reduction -> Z expectations ----------
    // D layout: lane holds basis n = p, VGPR r holds patch m = 8*hi + r.
    // z_q = sum_n (-1)^{n_q} |amp_n|^2  ==  WHT coefficient at h = 8>>q.
    #pragma unroll
    for (int r = 0; r < 8; ++r) {
        float x = accR[r] * accR[r] + accI[r] * accI[r];
        #pragma unroll
        for (int bit = 0; bit < 4; ++bit) {     // xor masks 1,2,4,8 stay within each 16-lane half
            float o = __shfl_xor(x, 1 << bit, 32);
            x = o + (((p >> bit) & 1) ? -x : x);
        }
        int q = -1;
        if      (p == 8) q = 0;
        else if (p == 4) q = 1;
        else if (p == 2) q = 2;
        else if (p == 1) q = 3;
        if (q >= 0) {
            const int gm = tile * 16 + hi * 8 + r;
            if (gm < nPatches) {
                const unsigned bm  = (unsigned)gm / PPB;
                const unsigned rmm = (unsigned)gm % PPB;
                // reference's reshape keeps z's flat order: b*15876 + patch*4 + q
                out[(size_t)bm * OUTB + rmm * 4 + q] = x;
            }
        }
    }
}

extern "C" void kernel_launch(void* const* d_in, const int* in_sizes, int n_in,
                              void* d_out, int out_size, void* d_ws, size_t ws_size,
                              hipStream_t stream) {
    const float* X = (const float*)d_in[0];   // (B,1,128,128) f32
    const float* W = (const float*)d_in[1];   // (1,8) f32
    float* out = (float*)d_out;               // B*15876 f32, flat order b*15876 + patch*4 + q

    const int B        = in_sizes[0] / (IMG * IMG);
    const int nPatches = B * PPB;                   // 254016 for B=64
    const int nTiles   = (nPatches + 15) / 16;      // 15876
    const int blocks   = (nTiles + 3) / 4;          // 4 waves (tiles) per 128-thread block

    quanv2d_wmma_kernel<<<blocks, 128, 0, stream>>>(X, W, out, nTiles, nPatches);
}